// DAGCN_full_7705171329818
// MI455X (gfx1250) — compile-verified
//
#include <hip/hip_runtime.h>

// DAGCN fused pipeline for MI455X (gfx1250, wave32, WMMA fp32 16x16x4).
//
// B=32, T=24, N=512, C=64, O=64, K(cheb)=3.
//
// Algebraic folding:
//   supp2 = 2*diag(S)*S - I  =>  x_g only needs y = S @ x
//   out[b,t,n,:] = x_row*(W0-W2) + y_row*(W1 + d*W2) + bias,  d = 2*S[t,n,n]
//
// Workspace layout (floats): scores[24*512*512] | d[24*512] | y[32*24*512*64]
// Total ~120 MB.  All LDS operands laid out so each WMMA fragment is a single
// aligned ds_load_b64 into an even VGPR pair (strides == 4 mod 64 banks).

typedef float v2f __attribute__((ext_vector_type(2)));
typedef float v8f __attribute__((ext_vector_type(8)));

#define B_ 32
#define T_ 24
#define N_ 512
#define C_ 64
#define O_ 64

// ---------------------------------------------------------------------------
// Kernel 1: scores[t,n,:] = softmax(relu(E[t,n]*E[t,:])),  d[t,n]=2*scores[t,n,n]
// ---------------------------------------------------------------------------
__global__ __launch_bounds__(256)
void k_scores(const float* __restrict__ E, float* __restrict__ scores,
              float* __restrict__ dvec) {
  const int row = blockIdx.x;            // t*512 + n
  const int t = row >> 9;
  const int n = row & 511;
  const float* Et = E + t * N_;
  const float en = Et[n];
  const int tid = threadIdx.x;

  float v0 = fmaxf(en * Et[tid], 0.0f);
  float v1 = fmaxf(en * Et[tid + 256], 0.0f);

  __shared__ float red[256];
  red[tid] = fmaxf(v0, v1);
  __syncthreads();
  #pragma unroll
  for (int s = 128; s > 0; s >>= 1) {
    if (tid < s) red[tid] = fmaxf(red[tid], red[tid + s]);
    __syncthreads();
  }
  const float rowmax = red[0];
  __syncthreads();

  const float e0 = __expf(v0 - rowmax);
  const float e1 = __expf(v1 - rowmax);
  red[tid] = e0 + e1;
  __syncthreads();
  #pragma unroll
  for (int s = 128; s > 0; s >>= 1) {
    if (tid < s) red[tid] += red[tid + s];
    __syncthreads();
  }
  const float inv = 1.0f / red[0];

  float* out = scores + (size_t)row * N_;
  out[tid]       = e0 * inv;
  out[tid + 256] = e1 * inv;

  if (tid == (n & 255)) {                 // thread holding element s == n
    const float snn = (n < 256 ? e0 : e1) * inv;
    dvec[row] = 2.0f * snn;
  }
}

// ---------------------------------------------------------------------------
// Kernel 2: y[b,t] = scores[t] @ x[b,t]   (512x512 * 512x64, fp32 WMMA)
// grid = (2 M-strips of 256, B*T). 8 waves; wave w owns rows [w*32, w*32+32)
// as two 16-row strips sharing the 4 B fragments -> 8 WMMA per 6 ds_load_b64.
// LDS: A row-major stride 68, B transposed (Bt[c][k]) stride 68.
// ---------------------------------------------------------------------------
__global__ __launch_bounds__(256)
void k_spmm(const float* __restrict__ x, const float* __restrict__ scores,
            float* __restrict__ y) {
  const int m0 = blockIdx.x * 256;       // M strip
  const int bt = blockIdx.y;             // b*T + t
  const int t  = bt % T_;
  const float* St  = scores + (size_t)t * N_ * N_;
  const float* Xbt = x + (size_t)bt * N_ * C_;
  float*       Ybt = y + (size_t)bt * N_ * C_;

  __shared__ alignas(16) float As[256 * 68];
  __shared__ alignas(16) float Bt[64 * 68];

  const int tid  = threadIdx.x;
  const int wv   = tid >> 5;
  const int lane = tid & 31;
  const int l16  = lane & 15;
  const int hi   = lane >> 4;            // half-wave select

  v8f acc[2][4] = {};

  for (int k0 = 0; k0 < N_; k0 += 64) {
    __syncthreads();
    // stage A: 256 rows x 64 K-cols of scores (b128 load, b128 LDS store)
    #pragma unroll
    for (int j = 0; j < 16; ++j) {
      const int idx = tid + j * 256;     // 0..4095 float4 slots
      const int r  = idx >> 4;
      const int c4 = (idx & 15) << 2;
      const float4 v = *(const float4*)(St + (size_t)(m0 + r) * N_ + k0 + c4);
      float* dst = &As[r * 68 + c4];
      dst[0] = v.x; dst[1] = v.y; dst[2] = v.z; dst[3] = v.w;
    }
    // stage B transposed: Bt[c][k] = x[k0+k][c].  Task = (c, k4-group):
    // 4 coalesced b32 global loads, one b128 LDS store.
    #pragma unroll
    for (int j = 0; j < 4; ++j) {
      const int idx = tid + j * 256;     // 0..1023 tasks
      const int c  = idx & 63;
      const int k4 = (idx >> 6) << 2;
      float* dst = &Bt[c * 68 + k4];
      dst[0] = Xbt[(size_t)(k0 + k4 + 0) * C_ + c];
      dst[1] = Xbt[(size_t)(k0 + k4 + 1) * C_ + c];
      dst[2] = Xbt[(size_t)(k0 + k4 + 2) * C_ + c];
      dst[3] = Xbt[(size_t)(k0 + k4 + 3) * C_ + c];
    }
    __syncthreads();

    const int ar0 = (wv * 32 + l16) * 68 + 2 * hi;   // strip 0
    const int ar1 = ar0 + 16 * 68;                   // strip 1
    #pragma unroll
    for (int kk = 0; kk < 64; kk += 4) {
      // A fragments: 16x4, VGPR v holds K = kk + v + 2*hi  (one b64 each)
      const v2f a0 = *(const v2f*)&As[ar0 + kk];
      const v2f a1 = *(const v2f*)&As[ar1 + kk];
      #pragma unroll
      for (int nt = 0; nt < 4; ++nt) {
        const v2f b = *(const v2f*)&Bt[(nt * 16 + l16) * 68 + kk + 2 * hi];
        acc[0][nt] = __builtin_amdgcn_wmma_f32_16x16x4_f32(
            false, a0, false, b, (short)0, acc[0][nt], false, false);
        acc[1][nt] = __builtin_amdgcn_wmma_f32_16x16x4_f32(
            false, a1, false, b, (short)0, acc[1][nt], false, false);
      }
    }
  }

  // C/D layout: VGPR r holds M = r (lanes 0-15) and M = r+8 (lanes 16-31)
  const int mbase = m0 + wv * 32 + 8 * hi;
  #pragma unroll
  for (int mt = 0; mt < 2; ++mt) {
    #pragma unroll
    for (int r = 0; r < 8; ++r) {
      float* yr = Ybt + (size_t)(mbase + mt * 16 + r) * C_;
      yr[ 0 + l16] = acc[mt][0][r];
      yr[16 + l16] = acc[mt][1][r];
      yr[32 + l16] = acc[mt][2][r];
      yr[48 + l16] = acc[mt][3][r];
    }
  }
}

// ---------------------------------------------------------------------------
// Kernel 3: per (t,n) folded output GEMM 32x64x128.
//   A[32x128] = [ x[b,t,n,:] | y[b,t,n,:] ]      (row-major, stride 132)
//   Bt[64x128] = folded weights, o-major:  Bt[o][i] = (W0-W2 ; W1+d*W2)[i][o]
//   out[b,t,n,o] = (A@B)[b,o] + bias[t,n,o]
// 8 waves, one 16x16 tile each; 32 WMMAs per wave, 2 ds_load_b64 per WMMA.
// Weights are streamed with non-temporal loads (read-once, 603 MB).
// ---------------------------------------------------------------------------
__global__ __launch_bounds__(256)
void k_out(const float* __restrict__ x, const float* __restrict__ y,
           const float* __restrict__ W, const float* __restrict__ bias,
           const float* __restrict__ dvec, float* __restrict__ out) {
  const int tn = blockIdx.x;             // t*512 + n
  const int t = tn >> 9;
  const int n = tn & 511;
  const float d = dvec[tn];
  const float* Wk = W + (size_t)tn * 3 * (C_ * O_);   // [k][i][o]

  __shared__ alignas(16) float Bt[64 * 132];   // Bt[o][i], i = 0..127
  __shared__ alignas(16) float As[32 * 132];   // As[b][i]

  const int tid = threadIdx.x;

  // Build folded, transposed B.  Task = (o, i4-group): 8 coalesced NT global
  // loads + one b128 LDS store.  Branch is wave-uniform (i4 uniform per wave).
  #pragma unroll
  for (int j = 0; j < 8; ++j) {
    const int idx = tid + j * 256;       // 0..2047 tasks
    const int o  = idx & 63;
    const int i4 = (idx >> 6) << 2;      // 0,4,...,124
    float* dst = &Bt[o * 132 + i4];
    if (i4 < 64) {
      #pragma unroll
      for (int u = 0; u < 4; ++u) {
        const float w0 = __builtin_nontemporal_load(Wk + (i4 + u) * O_ + o);
        const float w2 = __builtin_nontemporal_load(Wk + 2 * C_ * O_ + (i4 + u) * O_ + o);
        dst[u] = w0 - w2;
      }
    } else {
      #pragma unroll
      for (int u = 0; u < 4; ++u) {
        const float w1 = __builtin_nontemporal_load(Wk + C_ * O_ + (i4 - 64 + u) * O_ + o);
        const float w2 = __builtin_nontemporal_load(Wk + 2 * C_ * O_ + (i4 - 64 + u) * O_ + o);
        dst[u] = w1 + d * w2;
      }
    }
  }

  // Build A: per b, columns 0..63 = x slice, 64..127 = y slice
  #pragma unroll
  for (int j = 0; j < 4; ++j) {
    const int idx = tid + j * 256;       // 0..1023 float4 slots
    const int b  = idx >> 5;             // 0..31
    const int c4 = idx & 31;
    const float* src = (c4 < 16) ? x : y;
    const int cc = (c4 & 15) << 2;
    const float4 v =
        *(const float4*)(src + (((size_t)b * T_ + t) * N_ + n) * C_ + cc);
    float* dst = &As[b * 132 + (c4 < 16 ? 0 : 64) + cc];
    dst[0] = v.x; dst[1] = v.y; dst[2] = v.z; dst[3] = v.w;
  }
  __syncthreads();

  const int wv   = tid >> 5;
  const int lane = tid & 31;
  const int l16  = lane & 15;
  const int hi   = lane >> 4;
  const int mt = (wv >> 2) * 16;         // 0 or 16 (b tile)
  const int nt = (wv & 3) * 16;          // 0/16/32/48 (o tile)

  v8f acc = {};
  const int abase = (mt + l16) * 132 + 2 * hi;
  const int bbase = (nt + l16) * 132 + 2 * hi;
  #pragma unroll
  for (int kk = 0; kk < 128; kk += 4) {
    const v2f a = *(const v2f*)&As[abase + kk];
    const v2f b = *(const v2f*)&Bt[bbase + kk];
    acc = __builtin_amdgcn_wmma_f32_16x16x4_f32(false, a, false, b,
                                                (short)0, acc, false, false);
  }

  const float bv = bias[(size_t)tn * O_ + nt + l16];
  #pragma unroll
  for (int r = 0; r < 8; ++r) {
    const int brow = mt + r + 8 * hi;    // batch index b
    __builtin_nontemporal_store(
        acc[r] + bv,
        out + (((size_t)brow * T_ + t) * N_ + n) * O_ + nt + l16);
  }
}

// ---------------------------------------------------------------------------
extern "C" void kernel_launch(void* const* d_in, const int* in_sizes, int n_in,
                              void* d_out, int out_size, void* d_ws, size_t ws_size,
                              hipStream_t stream) {
  const float* x    = (const float*)d_in[0];   // [B,T,N,C]
  const float* E    = (const float*)d_in[1];   // [T,N]
  const float* W    = (const float*)d_in[2];   // [T,N,K,C,O]
  const float* bias = (const float*)d_in[3];   // [T,N,O]
  float* out = (float*)d_out;                  // [B,T,N,O]

  float* scores = (float*)d_ws;                       // T*N*N floats
  float* dvec   = scores + (size_t)T_ * N_ * N_;      // T*N floats
  float* y      = dvec + (size_t)T_ * N_;             // B*T*N*C floats

  k_scores<<<dim3(T_ * N_), dim3(256), 0, stream>>>(E, scores, dvec);
  k_spmm<<<dim3(N_ / 256, B_ * T_), dim3(256), 0, stream>>>(x, scores, y);
  k_out<<<dim3(T_ * N_), dim3(256), 0, stream>>>(x, y, W, bias, dvec, out);
}